// HighFrequencyEncoder_79903571574981
// MI455X (gfx1250) — compile-verified
//
#include <hip/hip_runtime.h>
#include <hip/hip_bf16.h>

// ---------------------------------------------------------------------------
// HighFrequencyEncoder for MI455X (gfx1250, wave32).
// 3 layers of: h_hp = h - alpha * D^-1/2 A D^-1/2 h  (scatter-add, fp32 L2 atomics)
//              y    = h_hp @ W^T + b                 (V_WMMA_F32_16X16X4_F32)
//              h    = relu(batchnorm(y))             (layers 0,1 only)
// FP32 WMMA keeps reference precision; the problem is memory/atomic-bound,
// so there is no benefit to dropping precision for more matrix throughput.
// ---------------------------------------------------------------------------

typedef __attribute__((ext_vector_type(2))) float v2f;
typedef __attribute__((ext_vector_type(8))) float v8f;

#define NNODES 50000
#define NEDGES 800000
#define FDIM   128
#define MTILES (NNODES / 16)   // 3125, exact
#define ALPHA  0.5f
#define BN_EPS 1e-5f
#define GEMM_WAVES 4

// ---- degree: deg[row[e]] += 1 ---------------------------------------------
__global__ void deg_kernel(const int* __restrict__ row, float* __restrict__ deg, int E) {
    int e = blockIdx.x * blockDim.x + threadIdx.x;
    if (e < E) {
        __hip_atomic_fetch_add(&deg[row[e]], 1.0f,
                               __ATOMIC_RELAXED, __HIP_MEMORY_SCOPE_AGENT);
    }
}

// ---- dinv = deg > 0 ? deg^-1/2 : 0  (in place) ----------------------------
__global__ void dinv_kernel(float* __restrict__ deg, int n) {
    int i = blockIdx.x * blockDim.x + threadIdx.x;
    if (i < n) {
        float d = deg[i];
        deg[i] = (d > 0.0f) ? rsqrtf(d) : 0.0f;
    }
}

// ---- agg[row] += h[col] * dinv[row]*dinv[col] -----------------------------
// One 32-lane group per edge; each lane handles 4 of the 128 features.
__global__ void scatter_kernel(const int* __restrict__ row, const int* __restrict__ col,
                               const float* __restrict__ dinv,
                               const float* __restrict__ h,
                               float* __restrict__ agg, int E) {
    long long gid = (long long)blockIdx.x * blockDim.x + threadIdx.x;
    int e = (int)(gid >> 5);
    if (e >= E) return;
    int lane = (int)(gid & 31);
    int r = row[e];
    int c = col[e];
    float norm = dinv[r] * dinv[c];
    const float* hc = h + (size_t)c * FDIM;
    float* ar = agg + (size_t)r * FDIM;
#pragma unroll
    for (int j = 0; j < 4; ++j) {
        int f = lane + j * 32;
        __hip_atomic_fetch_add(&ar[f], hc[f] * norm,
                               __ATOMIC_RELAXED, __HIP_MEMORY_SCOPE_AGENT);
    }
}

// ---- out[m,n] = sum_k (h[m,k] - alpha*agg[m,k]) * W[n,k] + bias[n] --------
// One wave computes a 16x128 output slab via 8 accumulator tiles of
// V_WMMA_F32_16X16X4_F32. A fragment (16x4 fp32): lanes 0-15 hold rows,
// VGPR0/1 = K pair selected by lane half. B fragment (4x16): mirrored.
__global__ __launch_bounds__(GEMM_WAVES * 32)
void hp_gemm_kernel(const float* __restrict__ h, const float* __restrict__ agg,
                    const float* __restrict__ W, const float* __restrict__ bias,
                    float* __restrict__ out, int mtiles) {
    int wave = threadIdx.x >> 5;
    int lane = threadIdx.x & 31;
    int mtile = blockIdx.x * GEMM_WAVES + wave;
    if (mtile >= mtiles) return;          // whole-wave exit: EXEC stays all-1s

    int mm = lane & 15;                   // row within tile (A) / col (B,C)
    int kh = (lane >> 4) << 1;            // K sub-offset for this lane half

    const float* hrow = h   + (size_t)(mtile * 16 + mm) * FDIM + kh;
    const float* arow = agg + (size_t)(mtile * 16 + mm) * FDIM + kh;

    v8f acc[8];
#pragma unroll
    for (int j = 0; j < 8; ++j) acc[j] = (v8f)(0.0f);

#pragma unroll 4
    for (int k0 = 0; k0 < FDIM; k0 += 4) {
        v2f hv = *(const v2f*)(hrow + k0);
        v2f av = *(const v2f*)(arow + k0);
        v2f a;
        a.x = hv.x - ALPHA * av.x;        // fused high-pass residual
        a.y = hv.y - ALPHA * av.y;
#pragma unroll
        for (int j = 0; j < 8; ++j) {
            // B[k][n] = W[n][k]; lane reads W[(j*16+mm)*128 + k0 + kh .. +1]
            v2f b = *(const v2f*)(W + (size_t)(j * 16 + mm) * FDIM + k0 + kh);
            acc[j] = __builtin_amdgcn_wmma_f32_16x16x4_f32(
                false, a, false, b, (short)0, acc[j], false, false);
        }
    }

    // C/D layout: VGPR r holds row (r + 8*laneHalf), col = lane&15.
    int mo = (lane >> 4) * 8;
#pragma unroll
    for (int j = 0; j < 8; ++j) {
        float bb = bias[j * 16 + mm];
#pragma unroll
        for (int r = 0; r < 8; ++r) {
            out[(size_t)(mtile * 16 + mo + r) * FDIM + j * 16 + mm] = acc[j][r] + bb;
        }
    }
}

// ---- per-column sum & sumsq -----------------------------------------------
__global__ void bn_stats_kernel(const float* __restrict__ y,
                                float* __restrict__ stats, int n) {
    int c = threadIdx.x;                  // 128 threads: one column each
    float s = 0.0f, s2 = 0.0f;
    for (int r = blockIdx.x; r < n; r += gridDim.x) {
        float v = y[(size_t)r * FDIM + c];
        s += v;
        s2 += v * v;
    }
    __hip_atomic_fetch_add(&stats[c], s, __ATOMIC_RELAXED, __HIP_MEMORY_SCOPE_AGENT);
    __hip_atomic_fetch_add(&stats[FDIM + c], s2, __ATOMIC_RELAXED, __HIP_MEMORY_SCOPE_AGENT);
}

// ---- y = relu((y - m) * rsqrt(v + eps) * g + bt)  (in place) --------------
__global__ void bn_apply_kernel(float* __restrict__ y,
                                const float* __restrict__ stats,
                                const float* __restrict__ g,
                                const float* __restrict__ bt, int n) {
    long long i = (long long)blockIdx.x * blockDim.x + threadIdx.x;
    if (i >= (long long)n * FDIM) return;
    int c = (int)(i & (FDIM - 1));
    float invn = 1.0f / (float)n;
    float m = stats[c] * invn;
    float var = stats[FDIM + c] * invn - m * m;   // biased variance
    float o = (y[i] - m) * rsqrtf(var + BN_EPS) * g[c] + bt[c];
    y[i] = o > 0.0f ? o : 0.0f;
}

// ---------------------------------------------------------------------------
extern "C" void kernel_launch(void* const* d_in, const int* in_sizes, int n_in,
                              void* d_out, int out_size, void* d_ws, size_t ws_size,
                              hipStream_t stream) {
    const float* x   = (const float*)d_in[0];
    const int*   ei  = (const int*)d_in[1];     // [2, E]
    const float* W1  = (const float*)d_in[2];
    const float* b1  = (const float*)d_in[3];
    const float* W2  = (const float*)d_in[4];
    const float* b2  = (const float*)d_in[5];
    const float* W3  = (const float*)d_in[6];
    const float* b3  = (const float*)d_in[7];
    const float* g1  = (const float*)d_in[8];
    const float* bt1 = (const float*)d_in[9];
    const float* g2  = (const float*)d_in[10];
    const float* bt2 = (const float*)d_in[11];
    float* out = (float*)d_out;

    const int* row = ei;
    const int* col = ei + NEDGES;

    // workspace layout (fp32): dinv | agg | yA | yB | stats
    float* ws    = (float*)d_ws;
    float* dinv  = ws;                                   // NNODES (deg -> dinv in place)
    float* agg   = dinv + NNODES;                        // NNODES*FDIM
    float* yA    = agg + (size_t)NNODES * FDIM;          // NNODES*FDIM
    float* yB    = yA  + (size_t)NNODES * FDIM;          // NNODES*FDIM
    float* stats = yB  + (size_t)NNODES * FDIM;          // 2*FDIM

    // degree -> dinv (once; shared across all 3 layers)
    hipMemsetAsync(dinv, 0, NNODES * sizeof(float), stream);
    deg_kernel<<<(NEDGES + 255) / 256, 256, 0, stream>>>(row, dinv, NEDGES);
    dinv_kernel<<<(NNODES + 255) / 256, 256, 0, stream>>>(dinv, NNODES);

    long long sthreads = (long long)NEDGES * 32;
    int sgrid = (int)((sthreads + 255) / 256);
    int ggrid = (MTILES + GEMM_WAVES - 1) / GEMM_WAVES;
    long long ethreads = (long long)NNODES * FDIM;
    int egrid = (int)((ethreads + 255) / 256);

    // ---- layer 0: x -> yA ----
    hipMemsetAsync(agg, 0, (size_t)NNODES * FDIM * sizeof(float), stream);
    scatter_kernel<<<sgrid, 256, 0, stream>>>(row, col, dinv, x, agg, NEDGES);
    hp_gemm_kernel<<<ggrid, GEMM_WAVES * 32, 0, stream>>>(x, agg, W1, b1, yA, MTILES);
    hipMemsetAsync(stats, 0, 2 * FDIM * sizeof(float), stream);
    bn_stats_kernel<<<256, FDIM, 0, stream>>>(yA, stats, NNODES);
    bn_apply_kernel<<<egrid, 256, 0, stream>>>(yA, stats, g1, bt1, NNODES);

    // ---- layer 1: yA -> yB ----
    hipMemsetAsync(agg, 0, (size_t)NNODES * FDIM * sizeof(float), stream);
    scatter_kernel<<<sgrid, 256, 0, stream>>>(row, col, dinv, yA, agg, NEDGES);
    hp_gemm_kernel<<<ggrid, GEMM_WAVES * 32, 0, stream>>>(yA, agg, W2, b2, yB, MTILES);
    hipMemsetAsync(stats, 0, 2 * FDIM * sizeof(float), stream);
    bn_stats_kernel<<<256, FDIM, 0, stream>>>(yB, stats, NNODES);
    bn_apply_kernel<<<egrid, 256, 0, stream>>>(yB, stats, g2, bt2, NNODES);

    // ---- layer 2 (output): yB -> out, no BN/ReLU ----
    hipMemsetAsync(agg, 0, (size_t)NNODES * FDIM * sizeof(float), stream);
    scatter_kernel<<<sgrid, 256, 0, stream>>>(row, col, dinv, yB, agg, NEDGES);
    hp_gemm_kernel<<<ggrid, GEMM_WAVES * 32, 0, stream>>>(yB, agg, W3, b3, out, MTILES);
}